// Net_12077448037023
// MI455X (gfx1250) — compile-verified
//
#include <hip/hip_runtime.h>
#include <math.h>

// ---------------------------------------------------------------------------
// GCN forward for MI455X (gfx1250), wave32, WMMA f32 16x16x4.
//   layer1: h = x@W1+b1 ; a1 = segsum(norm * h[row] -> col) (incl self loops)
//   layer2: h2 = relu(a1)@W2+b2 ; a2 = segsum(...)
//   out   : log_softmax(a2@Wo + bo)
// Weights are pre-transposed (and N-padded) into workspace so the WMMA K-loop
// is branch-free with contiguous b64 loads for both A and B fragments.
// ---------------------------------------------------------------------------

typedef __attribute__((ext_vector_type(2))) float v2f;
typedef __attribute__((ext_vector_type(8))) float v8f;

#define N_NODES   50000
#define N_IN      512
#define HIDDEN    64
#define N_CLASSES 40
#define NC_PAD    48            // N_CLASSES padded to 3 WMMA col tiles
#define N_EDGES   800000

// ---------------------------------------------------------------- degree ---
__global__ void init_deg(float* __restrict__ deg) {
    int i = blockIdx.x * blockDim.x + threadIdx.x;
    if (i < N_NODES) deg[i] = 1.0f;              // self loop contributes 1
}

__global__ void scatter_deg(const long long* __restrict__ ei,
                            float* __restrict__ deg) {
    int e = blockIdx.x * blockDim.x + threadIdx.x;
    if (e < N_EDGES) atomicAdd(&deg[(int)ei[e]], 1.0f);   // row = ei[0][e]
}

__global__ void finalize_dis(float* __restrict__ deg) {
    int i = blockIdx.x * blockDim.x + threadIdx.x;
    if (i < N_NODES) deg[i] = rsqrtf(deg[i]);    // in-place: deg -> deg^-1/2
}

// -------------------------------------------- weight transpose + N-pad -----
// Wt[n*K + k] = (n < N) ? W[k*N + n] : 0     for n in [0, Npad)
__global__ void transpose_pad_w(const float* __restrict__ W,
                                float* __restrict__ Wt,
                                int K, int N, int Npad) {
    int tid = blockIdx.x * blockDim.x + threadIdx.x;
    if (tid < K * Npad) {
        int k = tid % K;
        int n = tid / K;
        Wt[(size_t)n * K + k] = (n < N) ? W[(size_t)k * N + n] : 0.0f;
    }
}

// -------------------------------------------------- WMMA f32 GEMM + bias ---
// C[M x N] = (RELU ? relu(A) : A)[M x K] @ W[K x N] + bias[N]
// Wt is the transposed, Npad-row weight. One wave per 16x16 output tile:
// blockIdx.x = M/16 block, wave id = N tile. Inner loop: 2 b64 loads + 1 WMMA.
template <bool RELU>
__global__ void gemm_wmma_f32(const float* __restrict__ A,
                              const float* __restrict__ Wt,
                              const float* __restrict__ bias,
                              float* __restrict__ C,
                              int K, int N) {
    const int lane = threadIdx.x & 31;
    const int wave = threadIdx.x >> 5;
    const int half = lane >> 4;     // 0: K pair {0,1}; 1: K pair {2,3}
    const int l16  = lane & 15;
    const int m0   = blockIdx.x * 16;
    const int n    = wave * 16 + l16;   // output column handled by this lane
    const int mA   = m0 + l16;          // A row loaded by this lane

    const float* __restrict__ Arow = A  + (size_t)mA * K + half * 2;
    const float* __restrict__ Wrow = Wt + (size_t)n  * K + half * 2;

    v8f acc = {};
    #pragma unroll 4
    for (int k0 = 0; k0 < K; k0 += 4) {
        // A 16x4 tile: lane (l16 = M) holds K = k0+2*half, +1 (contiguous b64)
        v2f av = *(const v2f*)(Arow + k0);
        if (RELU) { av.x = fmaxf(av.x, 0.0f); av.y = fmaxf(av.y, 0.0f); }
        // B 4x16 tile: lane (l16 = N) holds K = k0+2*half, +1 (contiguous b64)
        v2f bv = *(const v2f*)(Wrow + k0);
        // D = A*B + C  (fp32 in, fp32 accumulate: exact vs reference)
        acc = __builtin_amdgcn_wmma_f32_16x16x4_f32(
            /*neg_a=*/false, av, /*neg_b=*/false, bv,
            /*c_mod=*/(short)0, acc, /*reuse_a=*/false, /*reuse_b=*/false);
    }

    if (n < N) {
        const float b = bias[n];
        #pragma unroll
        for (int r = 0; r < 8; ++r) {
            const int mm = m0 + r + half * 8;   // C/D layout: M = r (+8 hi half)
            C[(size_t)mm * N + n] = acc[r] + b;
        }
    }
}

// ----------------------------------------------- aggregation (self loops) ---
// a[i][:] = dis[i]^2 * h[i][:]   (initializes accumulator; no memset needed)
__global__ void self_msg(const float* __restrict__ h,
                         const float* __restrict__ dis,
                         float* __restrict__ a) {
    int tid = blockIdx.x * blockDim.x + threadIdx.x;
    int i = tid >> 4;
    int c = (tid & 15) << 2;
    if (i < N_NODES) {
        float s = dis[i] * dis[i];
        const float4 hv = *(const float4*)(h + (size_t)i * HIDDEN + c);
        float4* ap = (float4*)(a + (size_t)i * HIDDEN + c);
        float4 r; r.x = s*hv.x; r.y = s*hv.y; r.z = s*hv.z; r.w = s*hv.w;
        *ap = r;
    }
}

// a[col][:] += dis[row]*dis[col] * h[row][:]   for each of 800K edges.
// h and a (12.8 MB each) live in the 192 MB L2, so the random gather/scatter
// stays on-chip. One thread per (edge, 4-feature chunk).
__global__ void edge_agg(const long long* __restrict__ ei,
                         const float* __restrict__ h,
                         const float* __restrict__ dis,
                         float* __restrict__ a) {
    long long tid = (long long)blockIdx.x * blockDim.x + threadIdx.x;
    int e = (int)(tid >> 4);
    int c = ((int)tid & 15) << 2;
    if (e < N_EDGES) {
        int r  = (int)ei[e];             // source (row)
        int cl = (int)ei[e + N_EDGES];   // target (col)
        float nrm = dis[r] * dis[cl];
        const float4 hv = *(const float4*)(h + (size_t)r * HIDDEN + c);
        float* ap = a + (size_t)cl * HIDDEN + c;
        atomicAdd(ap + 0, nrm * hv.x);
        atomicAdd(ap + 1, nrm * hv.y);
        atomicAdd(ap + 2, nrm * hv.z);
        atomicAdd(ap + 3, nrm * hv.w);
    }
}

// ------------------------------------------------------------ log_softmax ---
__global__ void log_softmax_rows(float* __restrict__ out) {
    int i = blockIdx.x * blockDim.x + threadIdx.x;
    if (i < N_NODES) {
        float* p = out + (size_t)i * N_CLASSES;
        float m = -INFINITY;
        #pragma unroll
        for (int j = 0; j < N_CLASSES; ++j) m = fmaxf(m, p[j]);
        float s = 0.0f;
        #pragma unroll
        for (int j = 0; j < N_CLASSES; ++j) s += expf(p[j] - m);
        float ls = logf(s) + m;
        #pragma unroll
        for (int j = 0; j < N_CLASSES; ++j) p[j] = p[j] - ls;
    }
}

// ------------------------------------------------------------------ launch --
extern "C" void kernel_launch(void* const* d_in, const int* in_sizes, int n_in,
                              void* d_out, int out_size, void* d_ws, size_t ws_size,
                              hipStream_t stream) {
    const float*     x  = (const float*)d_in[0];
    const float*     W1 = (const float*)d_in[1];
    const float*     b1 = (const float*)d_in[2];
    const float*     W2 = (const float*)d_in[3];
    const float*     b2 = (const float*)d_in[4];
    const float*     Wo = (const float*)d_in[5];
    const float*     bo = (const float*)d_in[6];
    const long long* ei = (const long long*)d_in[7];
    float* out = (float*)d_out;

    // workspace layout
    char* ws = (char*)d_ws;
    size_t off = 0;
    float* h   = (float*)(ws + off); off += (size_t)N_NODES * HIDDEN * 4;   // 12.8 MB
    float* a   = (float*)(ws + off); off += (size_t)N_NODES * HIDDEN * 4;   // 12.8 MB
    float* dis = (float*)(ws + off); off += (size_t)N_NODES * 4;            // 0.2 MB
    float* W1t = (float*)(ws + off); off += (size_t)HIDDEN * N_IN * 4;      // 128 KB
    float* W2t = (float*)(ws + off); off += (size_t)HIDDEN * HIDDEN * 4;    // 16 KB
    float* Wot = (float*)(ws + off); off += (size_t)NC_PAD * HIDDEN * 4;    // 12 KB

    const int TPB = 256;
    const int nodeBlocks = (N_NODES + TPB - 1) / TPB;          // 196
    const int edgeBlocks = (N_EDGES + TPB - 1) / TPB;          // 3125
    const int nodeF4Blk  = (N_NODES * 16 + TPB - 1) / TPB;     // 3125
    const int edgeF4Blk  = (N_EDGES * 16 + TPB - 1) / TPB;     // 50000
    const int gemmBlocks = N_NODES / 16;                       // 3125 (exact)

    // degree / normalization
    init_deg    <<<nodeBlocks, TPB, 0, stream>>>(dis);
    scatter_deg <<<edgeBlocks, TPB, 0, stream>>>(ei, dis);
    finalize_dis<<<nodeBlocks, TPB, 0, stream>>>(dis);

    // weight prep: transpose (+ pad Wo to 48 columns)
    transpose_pad_w<<<(N_IN * HIDDEN   + TPB - 1) / TPB, TPB, 0, stream>>>(
        W1, W1t, N_IN,   HIDDEN,    HIDDEN);
    transpose_pad_w<<<(HIDDEN * HIDDEN + TPB - 1) / TPB, TPB, 0, stream>>>(
        W2, W2t, HIDDEN, HIDDEN,    HIDDEN);
    transpose_pad_w<<<(HIDDEN * NC_PAD + TPB - 1) / TPB, TPB, 0, stream>>>(
        Wo, Wot, HIDDEN, N_CLASSES, NC_PAD);

    // layer 1: h = x@W1+b1 ; a = aggregate(h)
    gemm_wmma_f32<false><<<gemmBlocks, 32 * (HIDDEN / 16), 0, stream>>>(
        x, W1t, b1, h, N_IN, HIDDEN);
    self_msg<<<nodeF4Blk, TPB, 0, stream>>>(h, dis, a);
    edge_agg<<<edgeF4Blk, TPB, 0, stream>>>(ei, h, dis, a);

    // layer 2: h = relu(a)@W2+b2 ; a = aggregate(h)   (buffers reused)
    gemm_wmma_f32<true><<<gemmBlocks, 32 * (HIDDEN / 16), 0, stream>>>(
        a, W2t, b2, h, HIDDEN, HIDDEN);
    self_msg<<<nodeF4Blk, TPB, 0, stream>>>(h, dis, a);
    edge_agg<<<edgeF4Blk, TPB, 0, stream>>>(ei, h, dis, a);

    // output: logits = a@Wo+bo (3 col tiles cover 48, stores guarded at 40)
    gemm_wmma_f32<false><<<gemmBlocks, 32 * (NC_PAD / 16), 0, stream>>>(
        a, Wot, bo, out, HIDDEN, N_CLASSES);
    log_softmax_rows<<<nodeBlocks, TPB, 0, stream>>>(out);
}